// WavePlaneField_59021440582180
// MI455X (gfx1250) — compile-verified
//
#include <hip/hip_runtime.h>

#define NPTS 524288
#define FEAT 32
#define RES  256

typedef unsigned int u32x4 __attribute__((ext_vector_type(4)));
typedef int          i32x8 __attribute__((ext_vector_type(8)));
typedef int          i32x4 __attribute__((ext_vector_type(4)));
typedef float        f32x2 __attribute__((ext_vector_type(2)));

// Per-lane sample-offset table: lane 0 = base sample (no offset),
// lanes 1..12 = the 12 jitter offsets, lanes 13..31 unused (zero).
__device__ __constant__ float OXL[32] = {
  0.f, -1.f,-0.5f,0.5f,1.f, 0.f, 0.f, 0.f,0.f, 0.5f, 0.5f,-0.5f,-0.5f,
  0.f,0.f,0.f,0.f,0.f,0.f,0.f,0.f,0.f,0.f,0.f,0.f,0.f,0.f,0.f,0.f,0.f,0.f,0.f};
__device__ __constant__ float OYL[32] = {
  0.f,  0.f, 0.f, 0.f,0.f,-1.f,-0.5f,0.5f,1.f, 0.5f,-0.5f, 0.5f,-0.5f,
  0.f,0.f,0.f,0.f,0.f,0.f,0.f,0.f,0.f,0.f,0.f,0.f,0.f,0.f,0.f,0.f,0.f,0.f,0.f};

// ---------------------------------------------------------------------------
// Bilinear setup. Clamps so x1 = x0+1 is ALWAYS in range: if floor(x)==W-1 we
// use x0=W-2, wx=1.0, which selects the same texel as the reference's
// (x1=min(x0+1,W-1), wx=0) path. Identical result, keeps corners contiguous.
// ---------------------------------------------------------------------------
__device__ __forceinline__ void bilinearSetup(float px, float py,
                                              int& x0, int& y0,
                                              float& wx, float& wy) {
  float x = (px + 1.0f) * (0.5f * (float)(RES - 1));
  float y = (py + 1.0f) * (0.5f * (float)(RES - 1));
  x = fminf(fmaxf(x, 0.0f), (float)(RES - 1));
  y = fminf(fmaxf(y, 0.0f), (float)(RES - 1));
  float xf = floorf(x), yf = floorf(y);
  x0 = min((int)xf, RES - 2);
  y0 = min((int)yf, RES - 2);
  wx = x - (float)x0;
  wy = y - (float)y0;
}

// ---------------------------------------------------------------------------
// TDM-based re-layout: plane[c][y][x] (fp32) -> planeT[y][x][c].
// One block handles (plane, y, 64-wide x chunk). The Tensor Data Mover pulls a
// 2D tile: 32 "lines" (channels, line stride 65536 elems) x 64 elems into LDS
// with +1 DWORD pad per 64-DWORD row (LDS row stride 65 -> conflict-free
// transposed reads), then coalesced stores emit the channel-contiguous layout.
// ---------------------------------------------------------------------------
__global__ __launch_bounds__(256) void relayoutKernel(
    const float* __restrict__ p0, const float* __restrict__ p1,
    const float* __restrict__ p2, float* __restrict__ ws) {
  __shared__ float tile[32 * 65];

  const int chunk = blockIdx.x;   // 0..3   (x0 = chunk*64)
  const int y     = blockIdx.y;   // 0..255
  const int plane = blockIdx.z;   // 0..2
  const float* src = (plane == 0) ? p0 : (plane == 1) ? p1 : p2;
  const int x0 = chunk * 64;

  if (threadIdx.x < 32) {  // wave 0 issues the single TDM op (EXEC ignored)
    unsigned long long ga = (unsigned long long)(src + ((size_t)y * RES + x0));
    unsigned lds_off = (unsigned)(size_t)(&tile[0]);  // low 32 bits = LDS byte offset

    u32x4 g0;
    g0.x = 1u;                                   // count=1, user mode, no gather
    g0.y = lds_off;                              // lds_addr
    g0.z = (unsigned)ga;                         // global_addr[31:0]
    g0.w = (unsigned)((ga >> 32) & 0x01FFFFFFu)  // global_addr[56:32]
           | 0x80000000u;                        // type=2 ("image")

    i32x8 g1;
    g1.s0 = (2 << 16)      // data_size = 4 bytes
          | (1 << 20)      // pad_enable
          | (5 << 22);     // pad_interval: 64 DWORDs ; pad_amount=0 -> 1 DWORD
    g1.s1 = 64 << 16;      // tensor_dim0 = 64 (low 16 bits at [63:48])
    g1.s2 = 32 << 16;      // tensor_dim1 = 32 (low 16 bits at [95:80])
    g1.s3 = 64 << 16;      // tile_dim0   = 64 ([127:112])
    g1.s4 = 32;            // tile_dim1   = 32 channels; tile_dim2 = 0
    g1.s5 = 65536;         // tensor_dim0_stride low 32 (RES*RES elems between channels)
    g1.s6 = 0;             // stride0 hi16 = 0, stride1 low16 = 0
    g1.s7 = 1;             // tensor_dim1_stride bits[47:16] -> stride1 = 65536
    i32x4 z4 = {0, 0, 0, 0};
    i32x8 z8 = {0, 0, 0, 0, 0, 0, 0, 0};

    __builtin_amdgcn_tensor_load_to_lds(g0, g1, z4, z4, z8, 0);
    __builtin_amdgcn_s_wait_tensorcnt(0);
  }
  __syncthreads();

  float* dst = ws + ((size_t)plane * RES * RES + (size_t)y * RES + x0) * FEAT;
  #pragma unroll
  for (int j = 0; j < 8; ++j) {
    int o  = j * 256 + threadIdx.x;   // 0..2047
    int c  = o & 31;                  // consecutive lanes -> consecutive banks
    int xl = o >> 5;
    dst[(size_t)xl * FEAT + c] = tile[c * 65 + xl];
  }
}

// ---------------------------------------------------------------------------
// Main kernel (transposed layout): one wave per point, lane = channel slot.
// Phase 1 (per pair): lane j computes sample j's row byte-offset and weights
// (wave-uniform work done once across lanes instead of 32x redundantly).
// Phase 2: unrolled consume loop broadcasts each sample's setup with
// v_readlane (uniform -> row address stays scalar, loads use saddr form),
// then one coalesced 256B float2 load per y-row covers both x-corners
// (lanes 0..15 -> x0 column, lanes 16..31 -> x1 column). Packed-f32 FMAs
// accumulate; half sums combined with __shfl_xor(...,16) after 13 samples.
// ---------------------------------------------------------------------------
__global__ __launch_bounds__(256) void fieldKernelT(
    const float* __restrict__ pts, const float* __restrict__ scales,
    const float* __restrict__ planeT, const float* __restrict__ aabb,
    float* __restrict__ out) {
  const int lane = threadIdx.x & 31;
  int pt = __builtin_amdgcn_readfirstlane((int)(blockIdx.x * 8 + (threadIdx.x >> 5)));

  float S[3], PN[3];
  #pragma unroll
  for (int i = 0; i < 3; ++i) {
    float p  = pts[3 * (size_t)pt + i];
    S[i]     = scales[3 * (size_t)pt + i];
    float a0 = aabb[i], a1 = aabb[3 + i];
    PN[i] = (p - a0) * (2.0f / (a1 - a0)) - 1.0f;
  }

  const float oxl = OXL[lane];   // this lane's sample-offset (samples 0..12)
  const float oyl = OYL[lane];
  const int Q[3] = {0, 0, 1}, R[3] = {1, 2, 2};

  const bool hi = lane >= 16;
  // Branchless x-half weight: ax = hi ? swx : 1-swx  ==  selSgn*swx + selOff
  const float selSgn = hi ? 1.0f : -1.0f;
  const float selOff = hi ? 0.0f : 1.0f;
  const int laneOff = 2 * lane;   // float offset into the 64-float row pair

  f32x2 acc = {1.0f, 1.0f};

  #pragma unroll
  for (int pr = 0; pr < 3; ++pr) {
    const float bx = PN[Q[pr]], by = PN[R[pr]];
    const float sx = S[Q[pr]],  sy = S[R[pr]];
    const float* base = planeT + (size_t)pr * RES * RES * FEAT;

    // ---- Phase 1: lane j owns sample j (lane 0 = un-offset base point) ----
    float px = fmaf(sx, oxl, bx);
    float py = fmaf(sy, oyl, by);
    int x0, y0; float wx, wy;
    bilinearSetup(px, py, x0, y0, wx, wy);
    int addrb = (y0 * RES + x0) * (int)(FEAT * sizeof(float)); // row0 byte offset
    int wxi = __float_as_int(wx);
    int wyi = __float_as_int(wy);

    // ---- Phase 2: broadcast + gather ----
    f32x2 sum = {0.0f, 0.0f};
    #pragma unroll
    for (int j = 0; j < 13; ++j) {
      int   sa  = __builtin_amdgcn_readlane(addrb, j);                 // uniform
      float swx = __int_as_float(__builtin_amdgcn_readlane(wxi, j));   // uniform
      float swy = __int_as_float(__builtin_amdgcn_readlane(wyi, j));   // uniform
      const float* row0 = (const float*)((const char*)base + sa) + laneOff;
      f32x2 v0 = *(const f32x2*)(row0);              // y0 row, both x corners
      f32x2 v1 = *(const f32x2*)(row0 + RES * FEAT); // y1 row
      float ax = fmaf(selSgn, swx, selOff);          // 1-wx (lo half) / wx (hi half)
      float w1 = ax * swy;
      float w0 = ax - w1;                            // ax*(1-swy)
      f32x2 w0v = {w0, w0}, w1v = {w1, w1};
      sum = __builtin_elementwise_fma(v1, w1v, sum); // candidates for v_pk_fma_f32
      sum = __builtin_elementwise_fma(v0, w0v, sum);
    }
    float sumx = sum.x + __shfl_xor(sum.x, 16, 32);  // combine x0/x1 lane halves
    float sumy = sum.y + __shfl_xor(sum.y, 16, 32);
    f32x2 mean = {sumx * (1.0f / 13.0f), sumy * (1.0f / 13.0f)};
    acc *= mean;
  }
  if (lane < 16) {
    *(f32x2*)(out + (size_t)pt * FEAT + laneOff) = acc;
  }
}

// ---------------------------------------------------------------------------
// Fallback (ws too small): gather straight from the [C][H][W] layout,
// lane = channel. Correct but uncoalesced; only used if ws_size < 24 MB.
// ---------------------------------------------------------------------------
__global__ __launch_bounds__(256) void fieldKernelDirect(
    const float* __restrict__ pts, const float* __restrict__ scales,
    const float* __restrict__ p0, const float* __restrict__ p1,
    const float* __restrict__ p2, const float* __restrict__ aabb,
    float* __restrict__ out) {
  const int lane = threadIdx.x & 31;  // channel
  int pt = __builtin_amdgcn_readfirstlane((int)(blockIdx.x * 8 + (threadIdx.x >> 5)));

  float S[3], PN[3];
  #pragma unroll
  for (int i = 0; i < 3; ++i) {
    float p  = pts[3 * (size_t)pt + i];
    S[i]     = scales[3 * (size_t)pt + i];
    float a0 = aabb[i], a1 = aabb[3 + i];
    PN[i] = (p - a0) * (2.0f / (a1 - a0)) - 1.0f;
  }

  const float OX[12] = {-1.f,-0.5f,0.5f,1.f, 0.f, 0.f, 0.f,0.f, 0.5f, 0.5f,-0.5f,-0.5f};
  const float OY[12] = { 0.f, 0.f, 0.f,0.f,-1.f,-0.5f,0.5f,1.f, 0.5f,-0.5f, 0.5f,-0.5f};
  const int Q[3] = {0, 0, 1}, R[3] = {1, 2, 2};
  const float* PL[3] = {p0, p1, p2};

  float acc = 1.0f;
  #pragma unroll
  for (int pr = 0; pr < 3; ++pr) {
    const float bx = PN[Q[pr]], by = PN[R[pr]];
    const float sx = S[Q[pr]],  sy = S[R[pr]];
    const float* pl = PL[pr] + (size_t)lane * RES * RES;
    float sum = 0.0f;
    #pragma unroll
    for (int j = 0; j < 13; ++j) {
      float px = (j == 0) ? bx : fmaf(sx, OX[j - 1], bx);
      float py = (j == 0) ? by : fmaf(sy, OY[j - 1], by);
      int x0, y0; float wx, wy;
      bilinearSetup(px, py, x0, y0, wx, wy);
      const float* r0 = pl + (size_t)y0 * RES + x0;
      float f00 = r0[0], f10 = r0[1], f01 = r0[RES], f11 = r0[RES + 1];
      float v0 = fmaf(f10 - f00, wx, f00);
      float v1 = fmaf(f11 - f01, wx, f01);
      sum += fmaf(v1 - v0, wy, v0);
    }
    acc *= sum * (1.0f / 13.0f);
  }
  out[(size_t)pt * FEAT + lane] = acc;
}

extern "C" void kernel_launch(void* const* d_in, const int* in_sizes, int n_in,
                              void* d_out, int out_size, void* d_ws, size_t ws_size,
                              hipStream_t stream) {
  const float* pts    = (const float*)d_in[0];
  // d_in[1] = timestamps, unused by the reference
  const float* scales = (const float*)d_in[2];
  const float* p0     = (const float*)d_in[3];
  const float* p1     = (const float*)d_in[4];
  const float* p2     = (const float*)d_in[5];
  const float* aabb   = (const float*)d_in[6];
  float* out          = (float*)d_out;

  const size_t need = (size_t)3 * RES * RES * FEAT * sizeof(float);  // 24 MB
  if (ws_size >= need) {
    float* planeT = (float*)d_ws;
    dim3 tg(RES / 64, RES, 3);  // (chunk, y, plane)
    relayoutKernel<<<tg, 256, 0, stream>>>(p0, p1, p2, planeT);
    fieldKernelT<<<NPTS / 8, 256, 0, stream>>>(pts, scales, planeT, aabb, out);
  } else {
    fieldKernelDirect<<<NPTS / 8, 256, 0, stream>>>(pts, scales, p0, p1, p2, aabb, out);
  }
}